// GaussianVoxelizer_9689446220347
// MI455X (gfx1250) — compile-verified
//
#include <hip/hip_runtime.h>
#include <hip/hip_bf16.h>

#define GAUSS 32768
#define NFEAT 8
#define GX 200
#define GY 200
#define GZ 16
#define NVOX (GX * GY * GZ)

__device__ __constant__ const float kLOX = -40.0f;
__device__ __constant__ const float kLOY = -40.0f;
__device__ __constant__ const float kLOZ = -1.0f;
__device__ __constant__ const float kHIX = 40.0f;
__device__ __constant__ const float kHIY = 40.0f;
__device__ __constant__ const float kHIZ = 5.4f;
__device__ __constant__ const float kVOX = 0.4f;
__device__ __constant__ const float kEPS = 1e-6f;

typedef float v2f __attribute__((ext_vector_type(2)));
typedef float v8f __attribute__((ext_vector_type(8)));

// Scatter the 16x16 WMMA D-tile (dens ⊗ feat outer product) into the feature
// accumulator. D layout (wave32, f32 16x16): VGPR r, lane l -> row M = r + 8*(l>=16),
// col N = l % 16. Columns N >= NFEAT are zero-padding (inactive lanes).
// base_owner selects which lanes own the voxel flat-indices for this tile.
__device__ __forceinline__ void scatter_tile(const v8f d, int flat, int base_owner,
                                             float* __restrict__ gfeat, int lane) {
  const int n = lane & 15;            // feature column this lane holds
  const bool active = (n < NFEAT);
  const int mbase = (lane >> 4) << 3; // lanes 16..31 hold rows M = 8..15
#pragma unroll
  for (int r = 0; r < 8; ++r) {
    const int owner = base_owner + mbase + r;      // lane holding flat index of voxel M
    const int fl = __shfl(flat, owner, 32);        // ds_bpermute broadcast
    const float val = d[r];
    if (active && val != 0.0f) {
      atomicAdd(gfeat + fl * NFEAT + n, val);
    }
  }
}

__global__ void __launch_bounds__(256) splat_kernel(
    const float* __restrict__ means3d, const float* __restrict__ opac,
    const float* __restrict__ cov, const float* __restrict__ featin,
    float* __restrict__ gdens, float* __restrict__ gfeat) {
  const int lane = threadIdx.x & 31;
  const int g = blockIdx.x * (blockDim.x >> 5) + (threadIdx.x >> 5);  // wave-per-gaussian
  if (g >= GAUSS) return;  // uniform per wave

  // CDNA5 global_prefetch_b8: pull next gaussian's descriptor data toward L2/L0.
  if (g + 8 < GAUSS) {
    __builtin_prefetch(cov + (size_t)(g + 8) * 9, 0, 0);
    __builtin_prefetch(means3d + (size_t)(g + 8) * 3, 0, 0);
  }

  // Per-gaussian (wave-uniform) setup.
  const float mx = means3d[g * 3 + 0], my = means3d[g * 3 + 1], mz = means3d[g * 3 + 2];
  const float c00 = cov[g * 9 + 0], c01 = cov[g * 9 + 1], c02 = cov[g * 9 + 2];
  const float c11 = cov[g * 9 + 4], c12 = cov[g * 9 + 5], c22 = cov[g * 9 + 8];
  const float op = opac[g];

  const float sx = 3.0f * sqrtf(c00), sy = 3.0f * sqrtf(c11), sz = 3.0f * sqrtf(c22);
  const float blox = mx - sx, bloy = my - sy, bloz = mz - sz;
  const float bhix = mx + sx, bhiy = my + sy, bhiz = mz + sz;
  const bool valid = (bhix > kLOX) && (bhiy > kLOY) && (bhiz > kLOZ) &&
                     (blox < kHIX) && (bloy < kHIY) && (bloz < kHIZ);
  if (!valid) return;  // uniform per wave

  const int ilox = min((int)((fminf(fmaxf(blox, kLOX), kHIX) - kLOX) / kVOX), GX - 1);
  const int iloy = min((int)((fminf(fmaxf(bloy, kLOY), kHIY) - kLOY) / kVOX), GY - 1);
  const int iloz = min((int)((fminf(fmaxf(bloz, kLOZ), kHIZ) - kLOZ) / kVOX), GZ - 1);
  const int ihix = min((int)((fminf(fmaxf(bhix, kLOX), kHIX) - kLOX) / kVOX), GX - 1);
  const int ihiy = min((int)((fminf(fmaxf(bhiy, kLOY), kHIY) - kLOY) / kVOX), GY - 1);
  const int ihiz = min((int)((fminf(fmaxf(bhiz, kLOZ), kHIZ) - kLOZ) / kVOX), GZ - 1);

  // Symmetric 3x3 inverse via adjugate.
  const float A00c = c11 * c22 - c12 * c12;
  const float A01c = c02 * c12 - c01 * c22;
  const float A02c = c01 * c12 - c02 * c11;
  const float det = c00 * A00c + c01 * A01c + c02 * A02c;
  const float idet = 1.0f / det;
  const float a00 = A00c * idet, a01 = A01c * idet, a02 = A02c * idet;
  const float a11 = (c00 * c22 - c02 * c02) * idet;
  const float a12 = (c01 * c02 - c00 * c12) * idet;
  const float a22 = (c00 * c11 - c01 * c01) * idet;

  // B matrix for WMMA: row K=0 holds features[0..7], everything else zero.
  v2f bmat;
  bmat.x = (lane < NFEAT) ? featin[g * NFEAT + lane] : 0.0f;
  bmat.y = 0.0f;

  // 512 voxels of the 8x8x8 window; 32 voxels per iteration (one per lane).
  for (int t = 0; t < 16; ++t) {
    const int v = t * 32 + lane;
    const int vxo = v >> 6, vyo = (v >> 3) & 7, vzo = v & 7;
    const int jx = ilox + vxo, jy = iloy + vyo, jz = iloz + vzo;
    const int ix = min(jx, GX - 1), iy = min(jy, GY - 1), iz = min(jz, GZ - 1);
    const bool inb = (jx <= ihix) && (jy <= ihiy) && (jz <= ihiz);

    const float dx = ((float)ix + 0.5f) * kVOX + kLOX - mx;
    const float dy = ((float)iy + 0.5f) * kVOX + kLOY - my;
    const float dz = ((float)iz + 0.5f) * kVOX + kLOZ - mz;
    const float maha = a00 * dx * dx + a11 * dy * dy + a22 * dz * dz +
                       2.0f * (a01 * dx * dy + a02 * dx * dz + a12 * dy * dz);
    const float dens = inb ? op * __expf(-0.5f * maha) : 0.0f;
    const int flat = (ix * GY + iy) * GZ + iz;

    if (dens > 0.0f) atomicAdd(gdens + flat, dens);  // adding 0 is a value no-op: skip

    // __any result is wave-uniform -> EXEC stays all-1s inside (WMMA requirement).
    if (__any(dens != 0.0f)) {
      // Tile 0: voxels in lanes 0..15. A column K=0 = dens, rest zero.
      v2f amat0;
      amat0.x = (lane < 16) ? dens : 0.0f;
      amat0.y = 0.0f;
      // Tile 1: voxels in lanes 16..31, moved down to lanes 0..15.
      const float densHi = __shfl(dens, (lane & 15) + 16, 32);
      v2f amat1;
      amat1.x = (lane < 16) ? densHi : 0.0f;
      amat1.y = 0.0f;

      v8f acc0 = {};
      acc0 = __builtin_amdgcn_wmma_f32_16x16x4_f32(false, amat0, false, bmat,
                                                   (short)0, acc0, false, false);
      scatter_tile(acc0, flat, /*base_owner=*/0, gfeat, lane);

      v8f acc1 = {};
      acc1 = __builtin_amdgcn_wmma_f32_16x16x4_f32(false, amat1, false, bmat,
                                                   (short)0, acc1, false, false);
      scatter_tile(acc1, flat, /*base_owner=*/16, gfeat, lane);
    }
  }
}

__global__ void __launch_bounds__(256) zero_kernel(float4* __restrict__ p, int n4) {
  const int i = blockIdx.x * blockDim.x + threadIdx.x;
  if (i < n4) p[i] = make_float4(0.0f, 0.0f, 0.0f, 0.0f);
}

__global__ void __launch_bounds__(256) normalize_kernel(const float* __restrict__ gdens,
                                                        float* __restrict__ gfeat) {
  const int i = blockIdx.x * blockDim.x + threadIdx.x;
  if (i >= NVOX) return;
  const float invd = 1.0f / fmaxf(gdens[i], kEPS);
  float4* fp = reinterpret_cast<float4*>(gfeat + (size_t)i * NFEAT);
  float4 f0 = fp[0], f1 = fp[1];
  f0.x *= invd; f0.y *= invd; f0.z *= invd; f0.w *= invd;
  f1.x *= invd; f1.y *= invd; f1.z *= invd; f1.w *= invd;
  fp[0] = f0;
  fp[1] = f1;
}

extern "C" void kernel_launch(void* const* d_in, const int* in_sizes, int n_in,
                              void* d_out, int out_size, void* d_ws, size_t ws_size,
                              hipStream_t stream) {
  const float* means3d = (const float*)d_in[0];
  const float* opac = (const float*)d_in[1];
  const float* cov = (const float*)d_in[2];
  const float* featin = (const float*)d_in[3];

  float* out = (float*)d_out;
  float* gdens = out;          // NVOX floats (raw accumulated density)
  float* gfeat = out + NVOX;   // NVOX * NFEAT floats (normalized at the end)

  // 1) Zero-init the L2-resident accumulators (d_out is poisoned by harness).
  const int n4 = (NVOX * (1 + NFEAT)) / 4;  // 5,760,000 floats -> 1,440,000 float4
  zero_kernel<<<(n4 + 255) / 256, 256, 0, stream>>>((float4*)out, n4);

  // 2) Scatter splat: one wave32 per gaussian, 8 waves per block.
  splat_kernel<<<GAUSS / 8, 256, 0, stream>>>(means3d, opac, cov, featin, gdens, gfeat);

  // 3) Normalize features by clipped density.
  normalize_kernel<<<(NVOX + 255) / 256, 256, 0, stream>>>(gdens, gfeat);
}